// ViltEmbeddings_13451837571648
// MI455X (gfx1250) — compile-verified
//
#include <hip/hip_runtime.h>
#include <hip/hip_bf16.h>

typedef float v2f __attribute__((ext_vector_type(2)));
typedef float v8f __attribute__((ext_vector_type(8)));

#define HID 768
#define SEQ 40
#define BATCH 64
#define PD 12
#define NPATCH 144
#define TOKENS 185          // SEQ + 1 (cls) + 144 patches
#define KDIM 3072           // 3*32*32
#define IMGSZ 384

static __device__ __forceinline__ int imax(int a, int b) { return a > b ? a : b; }
static __device__ __forceinline__ int imin(int a, int b) { return a < b ? a : b; }

// ---------------- Kernel 1: text embeddings + LayerNorm + modality[0] ----------------
__global__ __launch_bounds__(256) void text_embed_kernel(
    const int* __restrict__ input_ids, const int* __restrict__ attn_mask,
    const int* __restrict__ token_type, const float* __restrict__ word_emb,
    const float* __restrict__ pos_emb, const float* __restrict__ tok_type_emb,
    const float* __restrict__ gamma, const float* __restrict__ beta,
    const float* __restrict__ modality_emb, float* __restrict__ out,
    float* __restrict__ mask_out)
{
  int bs = blockIdx.x;              // b*SEQ + s
  int b = bs / SEQ, s = bs - b * SEQ;
  int t = threadIdx.x;
  int id = input_ids[bs];
  int tt = token_type[bs];
  float e[3];
  float sum = 0.f, sq = 0.f;
#pragma unroll
  for (int j = 0; j < 3; ++j) {
    int c = t * 3 + j;
    float v = word_emb[(size_t)id * HID + c] + tok_type_emb[tt * HID + c] +
              pos_emb[s * HID + c];
    e[j] = v; sum += v; sq += v * v;
  }
  __shared__ float2 red[256];
  red[t] = make_float2(sum, sq);
  __syncthreads();
  for (int off = 128; off > 0; off >>= 1) {
    if (t < off) { red[t].x += red[t + off].x; red[t].y += red[t + off].y; }
    __syncthreads();
  }
  float mu   = red[0].x * (1.0f / HID);
  float var  = red[0].y * (1.0f / HID) - mu * mu;
  float rstd = rsqrtf(var + 1e-12f);
  size_t orow = ((size_t)b * TOKENS + s) * HID;
#pragma unroll
  for (int j = 0; j < 3; ++j) {
    int c = t * 3 + j;
    out[orow + c] = (e[j] - mu) * rstd * gamma[c] + beta[c] + modality_emb[c];
  }
  if (t == 0) mask_out[b * TOKENS + s] = (float)attn_mask[bs];
}

// ---------------- Kernel 2: patch-grid mask, valid-first destination permutation ----
__global__ __launch_bounds__(160) void mask_sel_kernel(
    const int* __restrict__ pixel_mask, int* __restrict__ dst,
    int* __restrict__ xh, int* __restrict__ xw, float* __restrict__ mask_out)
{
  int b = blockIdx.x, t = threadIdx.x;
  __shared__ int xm[NPATCH];
  __shared__ int dsts[NPATCH];
  if (t < NPATCH) {
    int py = t / PD, px = t - (t / PD) * PD;
    xm[t] = pixel_mask[((size_t)b * IMGSZ + py * 32) * IMGSZ + px * 32];
  }
  __syncthreads();
  if (t == 0) {
    int h = 0, w = 0, nv = 0;
    for (int y = 0; y < PD; ++y) h += xm[y * PD];
    for (int x = 0; x < PD; ++x) w += xm[x];
    for (int p = 0; p < NPATCH; ++p) nv += xm[p];
    xh[b] = h; xw[b] = w;
    int cv = 0, ci = 0;
    for (int p = 0; p < NPATCH; ++p) {       // stable: valid first, both in original order
      if (xm[p]) dsts[p] = cv++;
      else       dsts[p] = nv + ci++;
    }
  }
  __syncthreads();
  if (t < NPATCH) {
    dst[b * NPATCH + t] = dsts[t];
    mask_out[b * TOKENS + SEQ + 1 + dsts[t]] = (float)xm[t];
  }
  if (t == 0) mask_out[b * TOKENS + SEQ] = 1.0f;
}

// ---------------- Kernel 3: im2col GEMM (f32 WMMA) + fused bias/pos/modality/scatter ----
__global__ __launch_bounds__(256) void patch_gemm_kernel(
    const float* __restrict__ pixel, const float* __restrict__ convw,
    const float* __restrict__ convb, const float* __restrict__ vpos,
    const float* __restrict__ modality_emb, const int* __restrict__ iti_p,
    const int* __restrict__ dst, const int* __restrict__ xh,
    const int* __restrict__ xw, float* __restrict__ out)
{
  __shared__ float As[64][36];      // [M=64][K=32] +4 pad
  __shared__ float Bs[64][36];      // [N=64][K=32] +4 pad  (N-major: frag = contiguous float2)
  int nBase = blockIdx.x * 64;
  int mBase = blockIdx.y * 64;
  int t = threadIdx.x, lane = t & 31, wave = t >> 5;
  int rBase = (wave & 3) << 4;      // wave's M sub-row: 0,16,32,48
  int cBase = (wave >> 2) << 5;     // wave's N sub-col: 0 or 32
  v8f acc0 = {}, acc1 = {};

  // loader mapping: thread -> (row r, 8-float quarter q)
  int r = t >> 2, q = t & 3;
  int m = mBase + r;
  int bImg = m / NPATCH, p = m - bImg * NPATCH;
  int ph = p / PD, pw = p - ph * PD;
  const float* Abase = pixel + (size_t)bImg * 3 * IMGSZ * IMGSZ +
                       (size_t)(ph * 32) * IMGSZ + pw * 32;
  const float* Bbase = convw + (size_t)(nBase + r) * KDIM;

  for (int kb = 0; kb < KDIM; kb += 32) {
    // k = c*1024 + py*32 + px ; a 32-k run is one contiguous pixel row segment
    int ci = kb >> 10;
    int py = (kb & 1023) >> 5;
    const float* asrc = Abase + (size_t)ci * (IMGSZ * IMGSZ) + py * IMGSZ + q * 8;
    float4 av0 = *(const float4*)asrc;
    float4 av1 = *(const float4*)(asrc + 4);
    const float4* bsrc = (const float4*)(Bbase + kb + q * 8);
    float4 bv0 = bsrc[0];
    float4 bv1 = bsrc[1];
    *(float4*)&As[r][q * 8]     = av0;
    *(float4*)&As[r][q * 8 + 4] = av1;
    *(float4*)&Bs[r][q * 8]     = bv0;
    *(float4*)&Bs[r][q * 8 + 4] = bv1;

    // prefetch next K-block's source lines while this block computes
    int kbn = kb + 32;
    if (kbn < KDIM) {
      int cn  = kbn >> 10;
      int pyn = (kbn & 1023) >> 5;
      __builtin_prefetch(Abase + (size_t)cn * (IMGSZ * IMGSZ) + pyn * IMGSZ + q * 8, 0, 1);
      __builtin_prefetch(Bbase + kbn + q * 8, 0, 1);
    }
    __syncthreads();

    // WMMA fragment addressing (ISA 16x4 A layout: lanes0-15 K=vgpr, lanes16-31 K=2+vgpr)
    int aRow = rBase + (lane & 15);
    int kOff = (lane >> 4) << 1;
    int c0 = cBase + (lane & 15);
#pragma unroll
    for (int kk = 0; kk < 32; kk += 4) {
      float2 a2  = *(const float2*)&As[aRow][kk + kOff];
      float2 b2a = *(const float2*)&Bs[c0][kk + kOff];
      float2 b2b = *(const float2*)&Bs[c0 + 16][kk + kOff];
      v2f a;  a.x  = a2.x;  a.y  = a2.y;
      v2f b0; b0.x = b2a.x; b0.y = b2a.y;
      v2f b1; b1.x = b2b.x; b1.y = b2b.y;
      acc0 = __builtin_amdgcn_wmma_f32_16x16x4_f32(false, a, false, b0,
                                                   (short)0, acc0, false, false);
      acc1 = __builtin_amdgcn_wmma_f32_16x16x4_f32(false, a, false, b1,
                                                   (short)0, acc1, false, false);
    }
    __syncthreads();
  }

  // fused epilogue: + conv bias + bilinear pos emb + modality, scatter to token slot
  int iti = *iti_p;
  const float* mod = modality_emb + iti * HID;
#pragma unroll
  for (int i = 0; i < 8; ++i) {
    int Mi = rBase + i + ((lane >> 4) << 3);     // C layout: vgpr i -> M = i (+8 for hi lanes)
    int m2 = mBase + Mi;
    int b2 = m2 / NPATCH, p2 = m2 - b2 * NPATCH;
    int ph2 = p2 / PD, pw2 = p2 - ph2 * PD;
    int tok = dst[b2 * NPATCH + p2];
    int h = xh[b2], w = xw[b2];
    float hf = (float)imax(h - 1, 1), wf = (float)imax(w - 1, 1);
    float ry = (float)ph2 * (float)(PD - 1) / hf;
    float rx = (float)pw2 * (float)(PD - 1) / wf;
    int y0 = imin(imax((int)floorf(ry), 0), PD - 1);
    int x0 = imin(imax((int)floorf(rx), 0), PD - 1);
    int y1 = imin(y0 + 1, PD - 1);
    int x1 = imin(x0 + 1, PD - 1);
    float wy = ry - (float)y0, wx = rx - (float)x0;
    float valid = (ph2 < h && pw2 < w) ? 1.0f : 0.0f;
    const float* vp00 = vpos + (size_t)(1 + y0 * PD + x0) * HID;
    const float* vp01 = vpos + (size_t)(1 + y0 * PD + x1) * HID;
    const float* vp10 = vpos + (size_t)(1 + y1 * PD + x0) * HID;
    const float* vp11 = vpos + (size_t)(1 + y1 * PD + x1) * HID;
    size_t orow = ((size_t)b2 * TOKENS + SEQ + 1 + tok) * HID;
    float accv[2] = { acc0[i], acc1[i] };
#pragma unroll
    for (int tix = 0; tix < 2; ++tix) {
      int col = nBase + cBase + (lane & 15) + tix * 16;
      float pos = (vp00[col] * (1.f - wy) * (1.f - wx) +
                   vp01[col] * (1.f - wy) * wx +
                   vp10[col] * wy * (1.f - wx) +
                   vp11[col] * wy * wx) * valid;
      out[orow + col] = accv[tix] + convb[col] + pos + mod[col];
    }
  }
}

// ---------------- Kernel 4: CLS row ----------------
__global__ __launch_bounds__(256) void cls_kernel(
    const float* __restrict__ cls_tok, const float* __restrict__ vpos,
    const float* __restrict__ modality_emb, const int* __restrict__ iti_p,
    float* __restrict__ out, float* __restrict__ mask_out)
{
  int b = blockIdx.x, t = threadIdx.x;
  int iti = *iti_p;
  size_t orow = ((size_t)b * TOKENS + SEQ) * HID;
#pragma unroll
  for (int j = 0; j < 3; ++j) {
    int c = t * 3 + j;
    out[orow + c] = cls_tok[c] + vpos[c] + modality_emb[iti * HID + c];
  }
  if (t == 0) mask_out[b * TOKENS + SEQ] = 1.0f;
}

extern "C" void kernel_launch(void* const* d_in, const int* in_sizes, int n_in,
                              void* d_out, int out_size, void* d_ws, size_t ws_size,
                              hipStream_t stream) {
  const int*   input_ids    = (const int*)  d_in[0];
  const int*   attn_mask    = (const int*)  d_in[1];
  const int*   token_type   = (const int*)  d_in[2];
  const float* pixel_values = (const float*)d_in[3];
  const int*   pixel_mask   = (const int*)  d_in[4];
  const float* word_emb     = (const float*)d_in[5];
  const float* pos_emb_text = (const float*)d_in[6];
  const float* tok_type_emb = (const float*)d_in[7];
  const float* ln_gamma     = (const float*)d_in[8];
  const float* ln_beta      = (const float*)d_in[9];
  const float* cls_token    = (const float*)d_in[10];
  const float* conv_w       = (const float*)d_in[11];
  const float* conv_b       = (const float*)d_in[12];
  const float* vis_pos_emb  = (const float*)d_in[13];
  const float* modality_emb = (const float*)d_in[14];
  const int*   iti          = (const int*)  d_in[15];

  float* out      = (float*)d_out;
  float* mask_out = out + (size_t)BATCH * TOKENS * HID;

  int* dst = (int*)d_ws;                 // BATCH*NPATCH ints
  int* xh  = dst + BATCH * NPATCH;       // BATCH
  int* xw  = xh + BATCH;                 // BATCH

  text_embed_kernel<<<BATCH * SEQ, 256, 0, stream>>>(
      input_ids, attn_mask, token_type, word_emb, pos_emb_text, tok_type_emb,
      ln_gamma, ln_beta, modality_emb, out, mask_out);

  mask_sel_kernel<<<BATCH, 160, 0, stream>>>(pixel_mask, dst, xh, xw, mask_out);

  patch_gemm_kernel<<<dim3(HID / 64, (BATCH * NPATCH) / 64), 256, 0, stream>>>(
      pixel_values, conv_w, conv_b, vis_pos_emb, modality_emb, iti,
      dst, xh, xw, out);

  cls_kernel<<<BATCH, 256, 0, stream>>>(cls_token, vis_pos_emb, modality_emb,
                                        iti, out, mask_out);
}